// Winner_27315992003224
// MI455X (gfx1250) — compile-verified
//
#include <hip/hip_runtime.h>

#define IMG_H 512
#define IMG_W 512
#define NSLICE 96
#define BAND 16
#define NBANDS (IMG_H / BAND)   // 32
#define HALO 20                 // BAND + 4 halo rows
#define GUARD 2
#define PITCH 528               // 512 + 4 guard + 12 pad -> 528 % 64 = 16 (bank friendly)
#define THREADS 256
#define TILES_PER_WAVE 4        // 8 waves * 4 tiles * 16 cols = 512 cols

typedef float v2f __attribute__((ext_vector_type(2)));
typedef float v8f __attribute__((ext_vector_type(8)));
typedef unsigned int u32x4 __attribute__((ext_vector_type(4)));
typedef int i32x4 __attribute__((ext_vector_type(4)));
typedef int i32x8 __attribute__((ext_vector_type(8)));

#if __has_builtin(__builtin_amdgcn_tensor_load_to_lds)
#define HAVE_TDM 1
#else
#define HAVE_TDM 0
#endif
#if __has_builtin(__builtin_amdgcn_wmma_f32_16x16x4_f32)
#define HAVE_WMMA_F32 1
#else
#define HAVE_WMMA_F32 0
#endif

struct HS {
  float s1, s2;
};

// Horizontal 5-tap sums of x and x^2 at halo row h, image column n.
// After inlining, infer-address-spaces recovers ds_load for the LDS pointer.
__device__ __forceinline__ HS hsum5(const float* ldsbuf, int h, int n) {
  const float* r = ldsbuf + h * PITCH + GUARD + n;
  const float x0 = r[-2], x1 = r[-1], x2 = r[0], x3 = r[1], x4 = r[2];
  HS o;
  o.s1 = ((x0 + x1) + (x2 + x3)) + x4;
  o.s2 = fmaf(x4, x4, fmaf(x3, x3, fmaf(x2, x2, fmaf(x1, x1, x0 * x0))));
  return o;
}

template <bool PASS2>
__global__ __launch_bounds__(THREADS) void wiener_kernel(
    const float* __restrict__ x, const float* __restrict__ noise_in,
    float* __restrict__ out) {
  __shared__ float ldsbuf[HALO * PITCH];  // 42,240 B

  const int band = blockIdx.x;
  const int slice = blockIdx.y;
  const int r0 = band * BAND;
  const int row_lo = r0 - 2;
  const int valid_lo = (row_lo > 0) ? row_lo : 0;
  const int vh = r0 + BAND + 2;
  const int valid_hi = (vh < IMG_H) ? vh : IMG_H;
  const int nvalid = valid_hi - valid_lo;
  const int lds_row0 = valid_lo - row_lo;  // 0..2
  const size_t slice_base = (size_t)slice * (IMG_H * IMG_W);

  // Zero guard columns (cols 0,1 and 514,515 of every halo row).
  for (int i = threadIdx.x; i < HALO * 4; i += THREADS) {
    const int r = i >> 2, c = i & 3;
    const int col = (c < 2) ? c : (GUARD + IMG_W + (c - 2));
    ldsbuf[r * PITCH + col] = 0.0f;
  }
  // Zero vertical halo rows that fall outside the image (zero-fill 'SAME').
  for (int r = 0; r < HALO; ++r) {
    if (r < lds_row0 || r >= lds_row0 + nvalid) {
      for (int c = threadIdx.x; c < IMG_W; c += THREADS)
        ldsbuf[r * PITCH + GUARD + c] = 0.0f;
    }
  }

#if HAVE_TDM
  // One TDM descriptor per block: iterate mode, 1 row (512 f32) per
  // iteration, LDS pitch 528 elements, nvalid iterations.
  if (threadIdx.x < 32) {
    const unsigned long long g =
        (unsigned long long)(size_t)(x + slice_base + (size_t)valid_lo * IMG_W);
    const unsigned int lds_byte =
        (unsigned int)(size_t)(&ldsbuf[lds_row0 * PITCH + GUARD]);

    u32x4 g0;
    g0[0] = 1u;                                  // count=1, user descriptor
    g0[1] = lds_byte;                            // lds_addr (bytes)
    g0[2] = (unsigned int)(g & 0xffffffffu);     // global_addr lo32
    g0[3] = (unsigned int)((g >> 32) & 0x1ffffffu) | (2u << 30);  // type=2

    i32x8 g1;
    g1[0] = (2 << 16) | (1 << 19);               // data_size=4B, iterate_en
    g1[1] = (IMG_W & 0xffff) << 16;              // tensor_dim0 lo16
    g1[2] = (IMG_W >> 16) | ((IMG_H & 0xffff) << 16);  // td0 hi | td1 lo
    g1[3] = (IMG_H >> 16) | (IMG_W << 16);       // td1 hi | tile_dim0=512
    g1[4] = 1;                                   // tile_dim1=1, tile_dim2=0
    g1[5] = IMG_W;                               // tensor_dim0_stride lo32
    g1[6] = (IMG_W & 0xffff) << 16;              // str0 hi=0 | str1 lo16
    g1[7] = 0;                                   // str1 hi32

    i32x4 g2;
    g2[0] = 0;                                   // tensor_dim2 (unused)
    g2[1] = PITCH;                               // lds_addr_increment (elems)
    g2[2] = IMG_W;                               // global_addr_increment lo32
    g2[3] = ((nvalid - 1) & 0xffff) << 16;       // iterate_count | ga_inc hi=0

    i32x4 g3 = {0, 0, 0, 0};
    i32x8 g4 = {0, 0, 0, 0, 0, 0, 0, 0};         // trailing group (6-arg form)
    __builtin_amdgcn_tensor_load_to_lds(g0, g1, g2, g3, g4, 0);
    __builtin_amdgcn_s_wait_tensorcnt(0);
  }
#else
  for (int i = threadIdx.x; i < nvalid * IMG_W; i += THREADS) {
    const int r = i / IMG_W, c = i - r * IMG_W;
    ldsbuf[(lds_row0 + r) * PITCH + GUARD + c] =
        x[slice_base + (size_t)(valid_lo + r) * IMG_W + c];
  }
#endif
  __syncthreads();

  const int lane = threadIdx.x & 31;
  const int wv = threadIdx.x >> 5;
  const int half = lane >> 4;   // lane half selects M/K sub-rows per ISA layout
  const int ln = lane & 15;

#if HAVE_WMMA_F32
  // Constant banded A (16x20): A[m][k] = 1 iff k in [m, m+4].
  // f32 16x16x4 A layout: lanes 0-15 M=lane, v0 holds K=4c (lo half) /
  // K=4c+2 (hi half), v1 holds K=4c+1 / 4c+3.
  v2f Aband[5];
#pragma unroll
  for (int c = 0; c < 5; ++c) {
    const int k0 = 4 * c + 2 * half;
    const int k1 = k0 + 1;
    Aband[c].x = (k0 >= ln && k0 <= ln + 4) ? 1.0f : 0.0f;
    Aband[c].y = (k1 >= ln && k1 <= ln + 4) ? 1.0f : 0.0f;
  }
#endif

  float noise = 0.0f;
  if (PASS2) noise = noise_in[slice] * (1.0f / (float)(IMG_H * IMG_W));

  float lvar_acc = 0.0f;

#pragma unroll
  for (int t = 0; t < TILES_PER_WAVE; ++t) {
    const int col0 = (wv * TILES_PER_WAVE + t) * 16;
    const int n = col0 + ln;  // this lane's image column (B/D layout)

    v8f s1v, s2v;
#if HAVE_WMMA_F32
    v8f acc1 = {0.f, 0.f, 0.f, 0.f, 0.f, 0.f, 0.f, 0.f};
    v8f acc2 = {0.f, 0.f, 0.f, 0.f, 0.f, 0.f, 0.f, 0.f};
#pragma unroll
    for (int c = 0; c < 5; ++c) {
      const int h0 = 4 * c + 2 * half;  // B rows for this lane's two elems
      const HS a = hsum5(ldsbuf, h0, n);
      const HS b = hsum5(ldsbuf, h0 + 1, n);
      v2f B1, B2;
      B1.x = a.s1;
      B1.y = b.s1;
      B2.x = a.s2;
      B2.y = b.s2;
      acc1 = __builtin_amdgcn_wmma_f32_16x16x4_f32(false, Aband[c], false, B1,
                                                   (short)0, acc1, false, false);
      acc2 = __builtin_amdgcn_wmma_f32_16x16x4_f32(false, Aband[c], false, B2,
                                                   (short)0, acc2, false, false);
    }
    s1v = acc1;
    s2v = acc2;
#else
    float hs1[HALO], hs2[HALO];
#pragma unroll
    for (int h = 0; h < HALO; ++h) {
      const HS a = hsum5(ldsbuf, h, n);
      hs1[h] = a.s1;
      hs2[h] = a.s2;
    }
#pragma unroll
    for (int v = 0; v < 8; ++v) {
      const int m = v + 8 * half;
      float a = 0.f, b = 0.f;
#pragma unroll
      for (int d = 0; d < 5; ++d) {
        a += hs1[m + d];
        b += hs2[m + d];
      }
      s1v[v] = a;
      s2v[v] = b;
    }
#endif

#pragma unroll
    for (int v = 0; v < 8; ++v) {
      const float lm = s1v[v] * (1.0f / 25.0f);
      const float lv = fmaf(-lm, lm, s2v[v] * (1.0f / 25.0f));
      if (PASS2) {
        const int m = v + 8 * half;  // C/D layout: row = v + 8*laneHalf
        const float xv = ldsbuf[(m + 2) * PITCH + GUARD + n];
        const float sv = fmaxf(lv, noise);
        const float f = 1.0f - noise / sv;
        const float res = fmaf(f, xv - lm, lm);
        out[slice_base + (size_t)(r0 + m) * IMG_W + n] = (lv < noise) ? lm : res;
      } else {
        lvar_acc += lv;
      }
    }
  }

  if (!PASS2) {
    // Deterministic block reduction of sum(lvar): wave shfl, then LDS.
    float s = lvar_acc;
#pragma unroll
    for (int off = 16; off > 0; off >>= 1) s += __shfl_down(s, off, 32);
    __syncthreads();  // tile reads done; safe to reuse LDS
    if (lane == 0) ldsbuf[wv] = s;
    __syncthreads();
    if (threadIdx.x == 0) {
      float tot = 0.f;
      for (int i = 0; i < 8; ++i) tot += ldsbuf[i];
      out[(size_t)slice * NBANDS + band] = tot;  // per-block partial
    }
  }
}

__global__ void noise_reduce_kernel(const float* __restrict__ partials,
                                    float* __restrict__ noise) {
  const int t = threadIdx.x;
  if (t < NSLICE) {
    float s = 0.f;
    for (int b = 0; b < NBANDS; ++b) s += partials[t * NBANDS + b];
    noise[t] = s;
  }
}

extern "C" void kernel_launch(void* const* d_in, const int* in_sizes, int n_in,
                              void* d_out, int out_size, void* d_ws,
                              size_t ws_size, hipStream_t stream) {
  (void)in_sizes; (void)n_in; (void)out_size; (void)ws_size;
  const float* x = (const float*)d_in[0];  // [32,3,512,512] f32
  float* out = (float*)d_out;
  float* partials = (float*)d_ws;             // 96*32 floats
  float* noise = partials + NSLICE * NBANDS;  // 96 floats

  dim3 grid(NBANDS, NSLICE);
  wiener_kernel<false><<<grid, THREADS, 0, stream>>>(x, nullptr, partials);
  noise_reduce_kernel<<<1, 128, 0, stream>>>(partials, noise);
  wiener_kernel<true><<<grid, THREADS, 0, stream>>>(x, noise, out);
}